// RegionalMeans_20761871909289
// MI455X (gfx1250) — compile-verified
//
#include <hip/hip_runtime.h>

typedef float v2f __attribute__((ext_vector_type(2)));
typedef float v4f __attribute__((ext_vector_type(4)));
typedef float v8f __attribute__((ext_vector_type(8)));

#define HW      65536        // H*W
#define CC      64           // channels
#define KK      64           // blocks (superpixels)
#define NFLOAT  ((size_t)CC * HW)          // 4,194,304 floats in x
#define NVEC    (NFLOAT / 4)               // float4 count
#define NSPLIT  32
#define SEG     (HW / NSPLIT)              // 2048 p-values per split

// ---------------------------------------------------------------- zero sums
__global__ __launch_bounds__(256) void zero_sums_kernel(float* __restrict__ sums) {
    sums[blockIdx.x * 256 + threadIdx.x] = 0.0f;   // grid 16 -> 4096 floats
}

// ------------------------------------------------------- per-region counts
__global__ __launch_bounds__(256) void count_kernel(const float* __restrict__ oh,
                                                    float* __restrict__ denom) {
    __shared__ float s[256];
    const int k = blockIdx.x;
    const v4f* row = (const v4f*)(oh + (size_t)k * HW);
    float c = 0.0f;
    for (int p = threadIdx.x; p < HW / 4; p += 256) {
        v4f v = row[p];
        c += v.x + v.y + v.z + v.w;
    }
    s[threadIdx.x] = c;
    __syncthreads();
    for (int off = 128; off > 0; off >>= 1) {
        if (threadIdx.x < off) s[threadIdx.x] += s[threadIdx.x + off];
        __syncthreads();
    }
    if (threadIdx.x == 0) {
        float cnt = s[0];
        denom[k] = cnt + (cnt == 0.0f ? 1.0f : 0.0f);
    }
}

// --------------------------------------------- 64x64 GEMM via f32 WMMA x4
// D[k][c] += sum_p oh[k][p] * x[c][p]; one wave per (16x16 tile, p-split)
__global__ __launch_bounds__(32) void wmma_partial_kernel(const float* __restrict__ oh,
                                                          const float* __restrict__ x,
                                                          float* __restrict__ sums) {
    const int tile  = blockIdx.x & 15;   // 4x4 grid of 16x16 tiles
    const int split = blockIdx.x >> 4;   // p-range split
    const int tk = tile >> 2;
    const int tc = tile & 3;

    const int lane = threadIdx.x;        // wave32
    const int half = lane >> 4;          // selects K pair {0,1} vs {2,3}
    const int m    = lane & 15;          // A row / B column within tile

    // A 16x4 layout: lanes0-15 VGPR0=K0,VGPR1=K1 ; lanes16-31 VGPR0=K2,VGPR1=K3
    const float* arow = oh + (size_t)(tk * 16 + m) * HW + 2 * half;
    // B 4x16 layout (symmetric striping over lane halves)
    const float* brow = x  + (size_t)(tc * 16 + m) * HW + 2 * half;

    v8f acc = {};
    const int p0 = split * SEG;
    for (int p = p0; p < p0 + SEG; p += 16) {
        v2f a0 = *(const v2f*)(arow + p);
        v2f b0 = *(const v2f*)(brow + p);
        v2f a1 = *(const v2f*)(arow + p + 4);
        v2f b1 = *(const v2f*)(brow + p + 4);
        v2f a2 = *(const v2f*)(arow + p + 8);
        v2f b2 = *(const v2f*)(brow + p + 8);
        v2f a3 = *(const v2f*)(arow + p + 12);
        v2f b3 = *(const v2f*)(brow + p + 12);
        acc = __builtin_amdgcn_wmma_f32_16x16x4_f32(false, a0, false, b0, (short)0, acc, false, false);
        acc = __builtin_amdgcn_wmma_f32_16x16x4_f32(false, a1, false, b1, (short)0, acc, false, false);
        acc = __builtin_amdgcn_wmma_f32_16x16x4_f32(false, a2, false, b2, (short)0, acc, false, false);
        acc = __builtin_amdgcn_wmma_f32_16x16x4_f32(false, a3, false, b3, (short)0, acc, false, false);
    }

    // C/D layout: VGPR r -> (M=r, lanes0-15) / (M=r+8, lanes16-31), N=lane%16
    const int gc = tc * 16 + m;
#pragma unroll
    for (int r = 0; r < 8; ++r) {
        const int gk = tk * 16 + r + half * 8;
        atomicAdd(&sums[gk * CC + gc], acc[r]);
    }
}

// ------------------------------------------------------------- finalize
__global__ __launch_bounds__(256) void finalize_kernel(const float* __restrict__ sums,
                                                       const float* __restrict__ denom,
                                                       float* __restrict__ out_means) {
    for (int i = threadIdx.x; i < KK * CC; i += 256)
        out_means[i] = sums[i] / denom[i >> 6];
}

// ------------------------------------- 64-fold broadcast (1.07 GB stream)
__global__ __launch_bounds__(256) void broadcast_kernel(const v4f* __restrict__ src,
                                                        v4f* __restrict__ dst) {
    const size_t i = (size_t)blockIdx.x * blockDim.x + threadIdx.x;  // < NVEC
    v4f v = src[i];
#pragma unroll
    for (int k = 0; k < KK; ++k)
        __builtin_nontemporal_store(v, &dst[(size_t)k * NVEC + i]);
}

extern "C" void kernel_launch(void* const* d_in, const int* in_sizes, int n_in,
                              void* d_out, int out_size, void* d_ws, size_t ws_size,
                              hipStream_t stream) {
    const float* x  = (const float*)d_in[0];   // [1,64,256,256]
    const float* oh = (const float*)d_in[1];   // [1,64,256,256] one-hot
    float* out = (float*)d_out;

    float* denom = (float*)d_ws;       // 64 floats
    float* sums  = denom + KK;         // 64*64 floats

    float* out_repeat = out;                               // [64,64,256,256]
    float* out_means  = out + (size_t)KK * CC * HW;        // [64,64]

    zero_sums_kernel<<<16, 256, 0, stream>>>(sums);
    count_kernel<<<KK, 256, 0, stream>>>(oh, denom);
    wmma_partial_kernel<<<16 * NSPLIT, 32, 0, stream>>>(oh, x, sums);
    finalize_kernel<<<1, 256, 0, stream>>>(sums, denom, out_means);
    broadcast_kernel<<<(unsigned)(NVEC / 256), 256, 0, stream>>>((const v4f*)x, (v4f*)out_repeat);
}